// ColorCurveRender_16303695856233
// MI455X (gfx1250) — compile-verified
//
#include <hip/hip_runtime.h>

typedef float v2f __attribute__((ext_vector_type(2)));
typedef float v8f __attribute__((ext_vector_type(8)));

// ---------------------------------------------------------------------------
// One wave32 per 16x16 output tile: D = A[16xK] * B[Kx16] + bias, optional relu.
// Uses V_WMMA_F32_16X16X4_F32 (full fp32 matrix path, no precision loss).
// A row-major (lda), B row-major (ldb).
// VGPR layouts per CDNA5 ISA 7.12.2:
//   A 16x4 f32 : lane L holds M=L&15, K = {0,1} (lanes 0-15) / {2,3} (lanes 16-31)
//   B 4x16 f32 : lane L holds N=L&15, K = {0,1} / {2,3} split the same way
//   D 16x16 f32: VGPR j holds M = j + 8*(L>=16), N = L&15
// Rows m >= Mv of D are never stored, so A rows >= Mv only need a VALID address,
// not zeros: clamp the row to Mv-1 (garbage lands only in unstored D rows).
// This keeps the inner loop branch-free: b64 + 2x b32 loads + one v_wmma.
// ---------------------------------------------------------------------------
__global__ __launch_bounds__(32)
void gemm_tile_wmma(const float* __restrict__ A, const float* __restrict__ Bw,
                    const float* __restrict__ bias, float* __restrict__ Out,
                    int K, int lda, int ldb, int ldo, int Mv, int doRelu)
{
    const int lane = threadIdx.x & 31;
    const int half = lane >> 4;       // 0 or 1
    const int mn   = lane & 15;       // A row / B,D column
    const int ncol = blockIdx.x * 16 + mn;
    const int kp   = half * 2;        // K-pair base for this lane half

    const int arow = (mn < Mv) ? mn : (Mv - 1);
    const float* ap = A  + (size_t)arow * lda + kp;   // +4 floats per step (8B aligned)
    const float* bp = Bw + (size_t)kp * ldb + ncol;   // +4*ldb floats per step
    const size_t bstep = (size_t)4 * ldb;

    v8f acc = {0.f,0.f,0.f,0.f,0.f,0.f,0.f,0.f};
    #pragma unroll 4
    for (int kk = 0; kk < K; kk += 4) {
        const v2f a = *(const v2f*)ap;
        v2f b;
        b.x = bp[0];
        b.y = bp[ldb];
        acc = __builtin_amdgcn_wmma_f32_16x16x4_f32(
            /*neg_a=*/false, a, /*neg_b=*/false, b,
            /*c_mod=*/(short)0, acc, /*reuse_a=*/false, /*reuse_b=*/false);
        ap += 4;
        bp += bstep;
    }

    const float bv = bias ? bias[ncol] : 0.f;
    #pragma unroll
    for (int j = 0; j < 8; ++j) {
        const int m = j + 8 * half;
        if (m < Mv) {
            float v = acc[j] + bv;
            if (doRelu) v = fmaxf(v, 0.f);
            Out[(size_t)m * ldo + ncol] = v;
        }
    }
}

// cat[b, 0:192]=outs[0], [192:384]=outs[1], [384:576]=outs[2]+outs[3]
__global__ void build_cat(const float* __restrict__ outs, float* __restrict__ cat)
{
    const int i = blockIdx.x * blockDim.x + threadIdx.x;  // 8*576 threads
    if (i >= 8 * 576) return;
    const int b = i / 576, j = i % 576;
    const int k = j / 192, q = j % 192;
    float v;
    if (k == 0)      v = outs[(0 * 8 + b) * 192 + q];
    else if (k == 1) v = outs[(1 * 8 + b) * 192 + q];
    else             v = outs[(2 * 8 + b) * 192 + q] + outs[(3 * 8 + b) * 192 + q];
    cat[i] = v;
}

// Per curve-row (set s in {all,f,b} x b x c): tbl[row][i] = (p_i*norm, norm/64*prefix_i)
// norm = 64 / (sum(p) + 1e-30). 72 blocks x 64 threads.
__global__ __launch_bounds__(64)
void build_tables(const float* __restrict__ p_all, const float* __restrict__ outs,
                  float* __restrict__ tbl)
{
    const int r = blockIdx.x;                 // 0..71
    const int s = r / 24, b = (r / 3) % 8, c = r % 3;
    const float* src = (s == 0) ? p_all : (outs + (size_t)(s - 1) * 8 * 192);
    src += b * 192 + c * 64;

    __shared__ float p[64];
    const int i = threadIdx.x;
    p[i] = src[i];
    __syncthreads();

    float sum = 0.f, pre = 0.f;
    for (int j = 0; j < 64; ++j) {
        const float pj = p[j];
        if (j < i) pre += pj;
        sum += pj;
    }
    const float norm = 64.f / (sum + 1e-30f);
    float2 e;
    e.x = p[i] * norm;                 // slope term
    e.y = pre * norm * 0.015625f;      // prefix term
    ((float2*)tbl)[(size_t)r * 64 + i] = e;
}

// Render: 64 blocks per batch image, 256 threads, float4 per thread.
// Piecewise-linear identity turns the 64-term einsum into one LDS float2
// lookup + one FMA per (curve,channel) evaluation -> HBM-bound.
__global__ __launch_bounds__(256)
void render(const float* __restrict__ x, const float* __restrict__ tbl,
            float* __restrict__ out)
{
    __shared__ float2 lds[9 * 64];            // [set][c][piece]
    const int b   = blockIdx.x >> 6;
    const int blk = blockIdx.x & 63;

    const float2* tbl2 = (const float2*)tbl;
    for (int t = threadIdx.x; t < 9 * 64; t += 256) {
        const int s = t / 192, rem = t % 192; // [b][c][i] contiguous inside a set
        lds[t] = tbl2[(size_t)(s * 24 + b * 3) * 64 + rem];
    }
    __syncthreads();

    const int v = blk * 256 + threadIdx.x;    // float4 index within H*W/4 = 16384
    const float4* xb = (const float4*)(x + (size_t)b * 3 * 65536);
    const float4 xr = xb[0 * 16384 + v];
    const float4 xg = xb[1 * 16384 + v];
    const float4 xc = xb[2 * 16384 + v];

    float4 gy;
    gy.x = 0.299f * xr.x + 0.587f * xg.x + 0.114f * xc.x;
    gy.y = 0.299f * xr.y + 0.587f * xg.y + 0.114f * xc.y;
    gy.z = 0.299f * xr.z + 0.587f * xg.z + 0.114f * xc.z;
    gy.w = 0.299f * xr.w + 0.587f * xg.w + 0.114f * xc.w;

    auto eval = [&](float val, int row) -> float {
        int k = (int)(val * 64.f);
        k = (k < 0) ? 0 : ((k > 63) ? 63 : k);
        const float2 e = lds[row * 64 + k];
        return fmaf(val - (float)k * 0.015625f, e.x, e.y);
    };

    float4* o = (float4*)out;
    #pragma unroll
    for (int s = 0; s < 3; ++s) {
        #pragma unroll
        for (int c = 0; c < 3; ++c) {
            const float4 src = (s == 0) ? (c == 0 ? xr : (c == 1 ? xg : xc)) : gy;
            const int row = s * 3 + c;
            float4 res;
            res.x = eval(src.x, row);
            res.y = eval(src.y, row);
            res.z = eval(src.z, row);
            res.w = eval(src.w, row);
            o[(size_t)s * 393216 + (size_t)(b * 3 + c) * 16384 + v] = res;
        }
    }
}

extern "C" void kernel_launch(void* const* d_in, const int* in_sizes, int n_in,
                              void* d_out, int out_size, void* d_ws, size_t ws_size,
                              hipStream_t stream)
{
    const float* x   = (const float*)d_in[0];
    const float* ff  = (const float*)d_in[1];
    const float* bf  = (const float*)d_in[2];
    const float* pw1 = (const float*)d_in[3];
    const float* pb1 = (const float*)d_in[4];
    const float* pw2 = (const float*)d_in[5];
    const float* pb2 = (const float*)d_in[6];
    const float* fw1 = (const float*)d_in[7];
    const float* fb1 = (const float*)d_in[8];
    const float* fw2 = (const float*)d_in[9];
    const float* fb2 = (const float*)d_in[10];

    float* ws    = (float*)d_ws;
    float* h1    = ws;            // [4][8][192]  = 6144
    float* outs  = ws + 6144;     // [4][8][192]  = 6144
    float* cat   = ws + 12288;    // [8][576]     = 4608
    float* hf    = ws + 16896;    // [8][192]     = 1536
    float* param = ws + 18432;    // [8][192]     = 1536
    float* tbl   = ws + 19968;    // [72][64][2]  = 9216

    // Projections layer 1: relu(feats @ w1 + b1), feats = [f,b,f,b]
    for (int k = 0; k < 4; ++k) {
        const float* A = (k & 1) ? bf : ff;
        gemm_tile_wmma<<<12, 32, 0, stream>>>(A, pw1 + (size_t)k * 512 * 192,
                                              pb1 + k * 192, h1 + (size_t)k * 8 * 192,
                                              512, 512, 192, 192, 8, 1);
    }
    // Projections layer 2: h @ w2 + b2
    for (int k = 0; k < 4; ++k) {
        gemm_tile_wmma<<<12, 32, 0, stream>>>(h1 + (size_t)k * 8 * 192,
                                              pw2 + (size_t)k * 192 * 192,
                                              pb2 + k * 192, outs + (size_t)k * 8 * 192,
                                              192, 192, 192, 192, 8, 0);
    }
    build_cat<<<18, 256, 0, stream>>>(outs, cat);
    // fc: relu(cat @ fc_w1 + fc_b1) @ fc_w2 + fc_b2
    gemm_tile_wmma<<<12, 32, 0, stream>>>(cat, fw1, fb1, hf, 576, 576, 192, 192, 8, 1);
    gemm_tile_wmma<<<12, 32, 0, stream>>>(hf, fw2, fb2, param, 192, 192, 192, 192, 8, 0);

    build_tables<<<72, 64, 0, stream>>>(param, outs, tbl);
    render<<<512, 256, 0, stream>>>(x, tbl, (float*)d_out);
}